// IncepGCNBlock_70531952935094
// MI455X (gfx1250) — compile-verified
//
#include <hip/hip_runtime.h>
#include <hip/hip_bf16.h>

#define D 64
#define EPS 1e-5f

typedef float v2f __attribute__((ext_vector_type(2)));
typedef float v8f __attribute__((ext_vector_type(8)));

// ---------------------------------------------------------------------------
// helpers
// ---------------------------------------------------------------------------
__device__ __forceinline__ unsigned mix32(unsigned v) {
    v ^= v >> 16; v *= 0x7feb352dU;
    v ^= v >> 15; v *= 0x846ca68bU;
    v ^= v >> 16;
    return v;
}

__global__ void fill_kernel(float* __restrict__ p, float v, int n) {
    int t = blockIdx.x * blockDim.x + threadIdx.x;
    if (t < n) p[t] = v;
}

// out[:, 0:64] = x
__global__ void copy_x_kernel(const float* __restrict__ x, float* __restrict__ out, int n) {
    int t = blockIdx.x * blockDim.x + threadIdx.x;
    if (t >= n * D) return;
    int i = t >> 6, c = t & 63;
    out[(size_t)i * 256 + c] = x[t];
}

// ---------------------------------------------------------------------------
// degree / normalization  (deg includes self-loop: initialized to 1.0)
// ---------------------------------------------------------------------------
__global__ void degree_kernel(const long long* __restrict__ ei, float* __restrict__ deg,
                              int E) {
    int e = blockIdx.x * blockDim.x + threadIdx.x;
    if (e >= E) return;
    int dst = (int)ei[(size_t)E + e];
    unsafeAtomicAdd(&deg[dst], 1.0f);
}

__global__ void rsqrt_kernel(float* __restrict__ d, int n) {
    int t = blockIdx.x * blockDim.x + threadIdx.x;
    if (t < n) d[t] = __frsqrt_rn(d[t]);   // deg >= 1 always (self loop)
}

// ---------------------------------------------------------------------------
// GEMM: Y[N,64] = H[N,64(stride)] @ W[64,64]   via V_WMMA_F32_16X16X4_F32
// one 16x16 output tile per wave, K=64 -> 16 chained WMMAs, W staged in LDS
// ---------------------------------------------------------------------------
__global__ void gemm64_wmma_kernel(const float* __restrict__ H, int hstride,
                                   const float* __restrict__ W,
                                   float* __restrict__ Y, int ntiles) {
    __shared__ float Wlds[D * D];           // 16 KB
    for (int idx = threadIdx.x; idx < D * D; idx += blockDim.x)
        Wlds[idx] = W[idx];
    __syncthreads();

    int wave = threadIdx.x >> 5;            // 8 waves / block
    int lane = threadIdx.x & 31;
    int tile = blockIdx.x * (blockDim.x >> 5) + wave;
    if (tile >= ntiles) return;

    int trow = tile >> 2;                   // 4 column tiles of 16
    int tcol = tile & 3;
    int i0 = trow * 16;
    int j0 = tcol * 16;

    int m    = lane & 15;                   // A row / B,C col within tile
    int koff = (lane >> 4) * 2;             // lanes 16-31 hold K+2,K+3

    const float* hrow = H + (size_t)(i0 + m) * hstride;
    const float* wcol = Wlds + j0 + m;

    v8f acc = {};
    #pragma unroll
    for (int k = 0; k < D; k += 4) {
        v2f a, b;
        a.x = hrow[k + koff];
        a.y = hrow[k + koff + 1];
        b.x = wcol[(k + koff) * D];
        b.y = wcol[(k + koff + 1) * D];
        acc = __builtin_amdgcn_wmma_f32_16x16x4_f32(
            /*neg_a=*/false, a, /*neg_b=*/false, b,
            /*c_mod=*/(short)0, acc, /*reuse_a=*/false, /*reuse_b=*/false);
    }

    // C/D layout: VGPR r -> M = r (lanes 0-15) or r+8 (lanes 16-31), N = lane&15
    int rbase = i0 + ((lane >> 4) ? 8 : 0);
    int col   = j0 + m;
    #pragma unroll
    for (int r = 0; r < 8; ++r)
        Y[(size_t)(rbase + r) * D + col] = acc[r];
}

// ---------------------------------------------------------------------------
// edge scatter: agg[dst] += y[src] * dis[src]*dis[dst]
// 16 lanes per edge, float4 per lane (64 channels)
// ---------------------------------------------------------------------------
__global__ void scatter_kernel(const long long* __restrict__ ei,
                               const float* __restrict__ dis,
                               const float* __restrict__ y,
                               float* __restrict__ agg, int E) {
    int t = blockIdx.x * blockDim.x + threadIdx.x;
    int e = t >> 4;
    if (e >= E) return;
    int c4  = (t & 15) * 4;
    int src = (int)ei[e];
    int dst = (int)ei[(size_t)E + e];
    float w = dis[src] * dis[dst];
    const float4 v = *reinterpret_cast<const float4*>(y + (size_t)src * D + c4);
    float* a = agg + (size_t)dst * D + c4;
    unsafeAtomicAdd(a + 0, v.x * w);
    unsafeAtomicAdd(a + 1, v.y * w);
    unsafeAtomicAdd(a + 2, v.z * w);
    unsafeAtomicAdd(a + 3, v.w * w);
}

// self-loop contribution + bias (no atomics needed: one writer per element)
__global__ void finalize_kernel(float* __restrict__ agg, const float* __restrict__ y,
                                const float* __restrict__ dis,
                                const float* __restrict__ bias, int n) {
    int t = blockIdx.x * blockDim.x + threadIdx.x;
    if (t >= n * D) return;
    int i = t >> 6, c = t & 63;
    float d = dis[i];
    agg[t] += y[t] * d * d + bias[c];
}

// ---------------------------------------------------------------------------
// BatchNorm stats: per-channel sum & sumsq (LDS reduce, then global atomics)
// block = 256 threads = 4 row-lanes x 64 channels
// ---------------------------------------------------------------------------
__global__ void bn_stats_kernel(const float* __restrict__ agg,
                                float* __restrict__ sums, float* __restrict__ sqs,
                                int n) {
    __shared__ float ls[256], lq[256];
    int c  = threadIdx.x & 63;
    int rg = threadIdx.x >> 6;
    float s = 0.f, q = 0.f;
    for (int i = blockIdx.x * 4 + rg; i < n; i += gridDim.x * 4) {
        float v = agg[(size_t)i * D + c];
        s += v;
        q += v * v;
    }
    ls[threadIdx.x] = s;
    lq[threadIdx.x] = q;
    __syncthreads();
    if (rg == 0) {
        s = ls[c] + ls[c + 64] + ls[c + 128] + ls[c + 192];
        q = lq[c] + lq[c + 64] + lq[c + 128] + lq[c + 192];
        unsafeAtomicAdd(&sums[c], s);
        unsafeAtomicAdd(&sqs[c], q);
    }
}

// ---------------------------------------------------------------------------
// BN apply + ReLU + dropout(p=0.5, deterministic hash) -> branch columns
// ---------------------------------------------------------------------------
__global__ void bn_apply_kernel(const float* __restrict__ agg,
                                const float* __restrict__ sums,
                                const float* __restrict__ sqs,
                                const float* __restrict__ gamma,
                                const float* __restrict__ beta,
                                float* __restrict__ hout,   // pre-offset to column base
                                int n, int layer) {
    int t = blockIdx.x * blockDim.x + threadIdx.x;
    if (t >= n * D) return;
    int i = t >> 6, c = t & 63;
    float invN = 1.0f / (float)n;
    float mu   = sums[c] * invN;
    float var  = sqs[c] * invN - mu * mu;
    float rs   = __frsqrt_rn(var + EPS);
    float v    = (agg[t] - mu) * rs * gamma[c] + beta[c];
    v = fmaxf(v, 0.0f);
    unsigned h = mix32((unsigned)t * 0x9E3779B9u + (unsigned)(layer + 1) * 0x85EBCA77u);
    v = (h & 0x80000000u) ? v * 2.0f : 0.0f;
    hout[(size_t)i * 256 + c] = v;
}

// ---------------------------------------------------------------------------
// orchestration
// ---------------------------------------------------------------------------
extern "C" void kernel_launch(void* const* d_in, const int* in_sizes, int n_in,
                              void* d_out, int out_size, void* d_ws, size_t ws_size,
                              hipStream_t stream) {
    const float*     x     = (const float*)d_in[0];
    const long long* ei    = (const long long*)d_in[1];   // int64 [2, E]
    const float*     Ws    = (const float*)d_in[2];       // [6,64,64]
    const float*     bs    = (const float*)d_in[3];       // [6,64]
    const float*     gs    = (const float*)d_in[4];       // [6,64]
    const float*     betas = (const float*)d_in[5];       // [6,64]
    float*           out   = (float*)d_out;               // [N,256]

    const int N = in_sizes[0] / D;
    const int E = in_sizes[1] / 2;

    float* wsf   = (float*)d_ws;
    float* dis   = wsf;                        // N
    float* y     = dis + N;                    // N*64
    float* agg   = y + (size_t)N * D;          // N*64
    float* stats = agg + (size_t)N * D;        // 128 (sums | sqs)

    const int TPB = 256;
    const int nElem  = N * D;
    const int gElem  = (nElem + TPB - 1) / TPB;
    const int gN     = (N + TPB - 1) / TPB;
    const int gE     = (E + TPB - 1) / TPB;
    const int gScat  = ((E * 16) + TPB - 1) / TPB;
    const int ntiles = (N / 16) * 4;
    const int gGemm  = (ntiles + 7) / 8;       // 8 waves (tiles) per block

    // degree -> dis = rsqrt(deg), deg seeded at 1.0 for the self loop
    fill_kernel<<<gN, TPB, 0, stream>>>(dis, 1.0f, N);
    degree_kernel<<<gE, TPB, 0, stream>>>(ei, dis, E);
    rsqrt_kernel<<<gN, TPB, 0, stream>>>(dis, N);

    // out[:, 0:64] = x
    copy_x_kernel<<<gElem, TPB, 0, stream>>>(x, out, N);

    // layer -> {output column base, first-layer-of-branch?}
    const int  cols[6]  = {64, 128, 128, 192, 192, 192};
    const bool first[6] = {true, true, false, true, false, false};

    for (int l = 0; l < 6; ++l) {
        const float* H  = first[l] ? x : (out + cols[l]);
        int hstride     = first[l] ? D : 256;

        gemm64_wmma_kernel<<<gGemm, TPB, 0, stream>>>(H, hstride, Ws + l * D * D, y, ntiles);
        fill_kernel<<<gElem, TPB, 0, stream>>>(agg, 0.0f, nElem);
        scatter_kernel<<<gScat, TPB, 0, stream>>>(ei, dis, y, agg, E);
        finalize_kernel<<<gElem, TPB, 0, stream>>>(agg, y, dis, bs + l * D, nElem / D);
        fill_kernel<<<1, 128, 0, stream>>>(stats, 0.0f, 128);
        bn_stats_kernel<<<256, TPB, 0, stream>>>(agg, stats, stats + D, N);
        bn_apply_kernel<<<gElem, TPB, 0, stream>>>(agg, stats, stats + D,
                                                   gs + l * D, betas + l * D,
                                                   out + cols[l], N, l);
    }
}